// TripleMatrixDecoder_25134148616789
// MI455X (gfx1250) — compile-verified
//
#include <hip/hip_runtime.h>

typedef float v2f __attribute__((ext_vector_type(2)));
typedef float v8f __attribute__((ext_vector_type(8)));
typedef unsigned int v4u __attribute__((ext_vector_type(4)));
typedef int v4i __attribute__((ext_vector_type(4)));
typedef int v8i __attribute__((ext_vector_type(8)));

#define B_N 8192
#define D_N 256
#define T_N 1000
#define HP  (D_N + 4)    // head/tail LDS row stride: 260 % 64 = 4 -> conflict-free column gathers
#define PS  (D_N + 16)   // TDM-padded M panel row stride: 272 (pad 16 dwords/row) -> half-wave rows 32 banks apart
#define PANEL_ROWS 16
#define PANEL_FLOATS (PANEL_ROWS * PS)

// ---------------- pass 0: zero counts & cursors ----------------
__global__ void tmd_init_k(int* __restrict__ counts, int* __restrict__ cursors) {
    int i = blockIdx.x * blockDim.x + threadIdx.x;
    if (i < T_N) { counts[i] = 0; cursors[i] = 0; }
}

// ---------------- pass 1: histogram of rel ----------------
__global__ void tmd_hist_k(const int* __restrict__ rel, int* __restrict__ counts) {
    int b = blockIdx.x * blockDim.x + threadIdx.x;
    if (b < B_N) atomicAdd(&counts[rel[b]], 1);
}

// ---------------- pass 2: exclusive prefix sum over T=1000 (single block) ----------------
__global__ void tmd_scan_k(const int* __restrict__ counts, int* __restrict__ offsets) {
    __shared__ int s[1024];
    int tid = threadIdx.x;
    int v = (tid < T_N) ? counts[tid] : 0;
    s[tid] = v;
    __syncthreads();
    for (int off = 1; off < 1024; off <<= 1) {
        int add = (tid >= off) ? s[tid - off] : 0;
        __syncthreads();
        s[tid] += add;
        __syncthreads();
    }
    if (tid < T_N) offsets[tid] = s[tid] - v;      // exclusive
    if (tid == T_N - 1) offsets[T_N] = s[tid];     // total == B_N
}

// ---------------- pass 3: scatter batch ids into bins ----------------
__global__ void tmd_scatter_k(const int* __restrict__ rel, const int* __restrict__ offsets,
                              int* __restrict__ cursors, int* __restrict__ binned) {
    int b = blockIdx.x * blockDim.x + threadIdx.x;
    if (b < B_N) {
        int t = rel[b];
        int p = atomicAdd(&cursors[t], 1);
        binned[offsets[t] + p] = b;
    }
}

// TDM: DMA one 16-row x 256-col f32 panel of M into LDS with 16-dword row padding.
// D# built per CDNA5 ISA sec 8.3-8.6. Tracked by TENSORcnt.
__device__ __forceinline__ void tdm_load_panel(const float* gsrc, unsigned lds_off_bytes) {
    unsigned long long ga = (unsigned long long)(uintptr_t)gsrc;
    v4u g0;
    g0[0] = 1u;                                            // count=1, user descriptor
    g0[1] = lds_off_bytes;                                 // lds_addr
    g0[2] = (unsigned)(ga & 0xFFFFFFFFu);                  // global_addr[31:0]
    g0[3] = (unsigned)((ga >> 32) & 0x01FFFFFFu) | (2u << 30); // global_addr[56:32] | type=2
    v8i g1;
    g1[0] = (2 << 16)        // data_size = 4 bytes
          | (1 << 20)        // pad_enable
          | (7 << 22)        // pad_interval: every 256 dwords (one 1KB row)
          | (15 << 25);      // pad_amount: 16 dwords -> row stride 272 words
    g1[1] = (D_N & 0xFFFF) << 16;          // tensor_dim0[15:0] in bits 63:48
    g1[2] = (D_N << 16);                   // tensor_dim0 hi=0 | tensor_dim1[15:0]=256
    g1[3] = (D_N << 16);                   // tensor_dim1 hi=0 | tile_dim0=256
    g1[4] = PANEL_ROWS;                    // tile_dim1=16, tile_dim2=0
    g1[5] = D_N;                           // tensor_dim0_stride lo32 = 256
    g1[6] = 0;                             // stride hi | tensor_dim1_stride lo
    g1[7] = 0;
    v4i z4 = {0, 0, 0, 0};
    v8i z8 = {0, 0, 0, 0, 0, 0, 0, 0};
    __builtin_amdgcn_tensor_load_to_lds(g0, g1, z4, z4, z8, 0);
}

// ---------------- pass 4: per-matrix grouped bilinear forms via WMMA + TDM ----------------
// One block per t. 256 threads = 8 waves. Wave w owns e-tiles {w, w+8}.
// M is streamed once per group as 16 double-buffered TDM panels; WMMA consumes from LDS.
__global__ __launch_bounds__(256) void tmd_compute_k(
    const float* __restrict__ head, const float* __restrict__ tail,
    const float* __restrict__ mats, const int* __restrict__ binned,
    const int* __restrict__ offsets, float* __restrict__ out) {

    int t = blockIdx.x;
    int start = offsets[t];
    int end   = offsets[t + 1];
    if (start == end) return;

    const float* __restrict__ Mp = mats + (size_t)t * (D_N * D_N);

    __shared__ float Hs[16 * HP];
    __shared__ float Ts[16 * HP];
    __shared__ float Mbuf[2 * PANEL_FLOATS];   // double-buffered TDM panels
    __shared__ float part_s[16 * 16];          // [e-tile][g] deterministic partials
    __shared__ int   bs[16];

    const unsigned mbuf_lds = (unsigned)(uintptr_t)(&Mbuf[0]);

    const int tid  = threadIdx.x;
    const int lane = tid & 31;
    const int wave = tid >> 5;
    const int half = lane >> 4;   // 0: lanes 0-15, 1: lanes 16-31
    const int col  = lane & 15;
    const int e0   = wave * 16;

    for (int g0i = start; g0i < end; g0i += 16) {
        if (tid < 16) bs[tid] = (g0i + tid < end) ? binned[g0i + tid] : -1;

        // kick off panel 0 DMA while we stage heads/tails
        if (wave == 0) tdm_load_panel(Mp, mbuf_lds);
        __syncthreads();

        // stage (zero-padded) head/tail rows of this group into LDS
        for (int i = tid; i < 16 * D_N; i += 256) {
            int row  = i >> 8;       // D_N == 256
            int dcol = i & (D_N - 1);
            int bi   = bs[row];
            float hv = 0.0f, tv = 0.0f;
            if (bi >= 0) {
                hv = head[(size_t)bi * D_N + dcol];
                tv = tail[(size_t)bi * D_N + dcol];
            }
            Hs[row * HP + dcol] = hv;
            Ts[row * HP + dcol] = tv;
        }

        v8f c0 = {};
        v8f c1 = {};

        for (int p = 0; p < 16; ++p) {
            if (wave == 0) {
                if (p < 15) {
                    tdm_load_panel(Mp + (size_t)(p + 1) * PANEL_ROWS * D_N,
                                   mbuf_lds + ((unsigned)((p + 1) & 1)) * PANEL_FLOATS * 4u);
                    __builtin_amdgcn_s_wait_tensorcnt(1);   // panel p complete, p+1 in flight
                } else {
                    __builtin_amdgcn_s_wait_tensorcnt(0);
                }
            }
            __syncthreads();   // panel p (and, first time, Hs/Ts) visible to all waves

            const float* __restrict__ Pb = &Mbuf[(p & 1) * PANEL_FLOATS];
            const int dbase = p * PANEL_ROWS;

            #pragma unroll
            for (int dl = 0; dl < PANEL_ROWS; dl += 4) {
                const int d  = dbase + dl;
                const int rb = dl + 2 * half;
                v2f a, b0, b1;
                // A fragment: 16x4 slab of H_G (shared by both e-tiles of this wave)
                a.x = Hs[col * HP + d + 2 * half + 0];
                a.y = Hs[col * HP + d + 2 * half + 1];
                // B fragments: 4x16 tiles of M from the padded LDS panel
                b0.x = Pb[(rb + 0) * PS + e0 + col];
                b0.y = Pb[(rb + 1) * PS + e0 + col];
                b1.x = Pb[(rb + 0) * PS + e0 + 128 + col];
                b1.y = Pb[(rb + 1) * PS + e0 + 128 + col];
                c0 = __builtin_amdgcn_wmma_f32_16x16x4_f32(
                        false, a, false, b0, (short)0, c0, false, false);
                c1 = __builtin_amdgcn_wmma_f32_16x16x4_f32(
                        false, a, false, b1, (short)0, c1, false, false);
            }
            __syncthreads();   // all waves done with buffer (p&1) before it is re-filled
        }

        // D layout: VGPR r of c holds Z[r + 8*half][e]; dot with tails + 16-lane reduce
        #pragma unroll
        for (int r = 0; r < 8; ++r) {
            int g = r + 8 * half;
            float v0 = c0[r] * Ts[g * HP + e0 + col];
            v0 += __shfl_xor(v0, 1);
            v0 += __shfl_xor(v0, 2);
            v0 += __shfl_xor(v0, 4);
            v0 += __shfl_xor(v0, 8);
            if (col == 0) part_s[wave * 16 + g] = v0;        // e-tile = wave
            float v1 = c1[r] * Ts[g * HP + e0 + 128 + col];
            v1 += __shfl_xor(v1, 1);
            v1 += __shfl_xor(v1, 2);
            v1 += __shfl_xor(v1, 4);
            v1 += __shfl_xor(v1, 8);
            if (col == 0) part_s[(wave + 8) * 16 + g] = v1;  // e-tile = wave+8
        }
        __syncthreads();

        if (tid < 16 && (g0i + tid) < end) {
            float acc = 0.0f;
            #pragma unroll
            for (int et = 0; et < 16; ++et) acc += part_s[et * 16 + tid];
            out[bs[tid]] = acc;
        }
        __syncthreads();
    }
}

extern "C" void kernel_launch(void* const* d_in, const int* in_sizes, int n_in,
                              void* d_out, int out_size, void* d_ws, size_t ws_size,
                              hipStream_t stream) {
    const int*   rel  = (const int*)d_in[0];
    const float* head = (const float*)d_in[1];
    const float* tail = (const float*)d_in[2];
    const float* mats = (const float*)d_in[3];
    float*       out  = (float*)d_out;

    int* ws      = (int*)d_ws;
    int* counts  = ws;                 // T
    int* offsets = ws + T_N;           // T+1
    int* cursors = ws + 2 * T_N + 1;   // T
    int* binned  = ws + 3 * T_N + 1;   // B

    tmd_init_k   <<<(T_N + 255) / 256, 256, 0, stream>>>(counts, cursors);
    tmd_hist_k   <<<(B_N + 255) / 256, 256, 0, stream>>>(rel, counts);
    tmd_scan_k   <<<1, 1024, 0, stream>>>(counts, offsets);
    tmd_scatter_k<<<(B_N + 255) / 256, 256, 0, stream>>>(rel, offsets, cursors, binned);
    tmd_compute_k<<<T_N, 256, 0, stream>>>(head, tail, mats, binned, offsets, out);
}